// SpGraphAttentionLayerV2_71889162600965
// MI455X (gfx1250) — compile-verified
//
#include <hip/hip_runtime.h>

#define NNODES 50000
#define NEDGES 800000
#define DFEAT  128
#define ALPHA  0.2f

typedef __attribute__((ext_vector_type(2))) float v2f;
typedef __attribute__((ext_vector_type(8))) float v8f;

// ---------------------------------------------------------------------------
// Kernel 1: h = X @ W  via V_WMMA_F32_16X16X4_F32 (exact fp32 matrix FMA).
// One wave32 computes a 16x128 strip of h: A-fragment loaded once per K-step,
// reused across 8 column tiles (8 v8f accumulators). 3125 row tiles total.
// A layout (16x4 f32): lane&15 = M; VGPR0 = K{0|2}, VGPR1 = K{1|3} by lane half.
// B layout (4x16 f32): lane&15 = N; same K striping.
// C/D layout (16x16 f32): VGPR r -> M = r + 8*(lane>>4), N = lane&15.
// ---------------------------------------------------------------------------
__global__ __launch_bounds__(256) void gat_gemm_wmma(const float* __restrict__ X,
                                                     const float* __restrict__ W,
                                                     float* __restrict__ H) {
    const int lane  = threadIdx.x & 31;
    const int wave  = threadIdx.x >> 5;
    const int rowTile = blockIdx.x * 8 + wave;
    if (rowTile * 16 >= NNODES) return;

    const int m0    = rowTile * 16;
    const int laneM = lane & 15;      // M for A, N for B/C/D
    const int half  = lane >> 4;

    v8f acc[8];
    const v8f zero = {0.f, 0.f, 0.f, 0.f, 0.f, 0.f, 0.f, 0.f};
#pragma unroll
    for (int t = 0; t < 8; ++t) acc[t] = zero;

    const float* Arow = X + (size_t)(m0 + laneM) * DFEAT;

    for (int k0 = 0; k0 < DFEAT; k0 += 4) {
        const int ka = k0 + 2 * half;           // this lane-half's K pair
        v2f afrag;
        afrag.x = Arow[ka];
        afrag.y = Arow[ka + 1];
#pragma unroll
        for (int t = 0; t < 8; ++t) {
            const int n = t * 16 + laneM;
            v2f bfrag;
            bfrag.x = W[(size_t)ka * DFEAT + n];
            bfrag.y = W[(size_t)(ka + 1) * DFEAT + n];
            acc[t] = __builtin_amdgcn_wmma_f32_16x16x4_f32(
                /*neg_a=*/false, afrag, /*neg_b=*/false, bfrag,
                /*c_mod=*/(short)0, acc[t], /*reuse_a=*/false, /*reuse_b=*/false);
        }
    }

#pragma unroll
    for (int t = 0; t < 8; ++t) {
#pragma unroll
        for (int r = 0; r < 8; ++r) {
            const int m = m0 + r + 8 * half;
            const int n = t * 16 + laneM;
            H[(size_t)m * DFEAT + n] = acc[t][r];
        }
    }
}

// ---------------------------------------------------------------------------
// Kernel 2: edge phase. One wave32 per edge; lane l owns features [4l,4l+4).
//   s  = leakyrelu(h[src]+h[dst]);  e = exp(dot(s, a))  (wave butterfly reduce)
//   out[src,:]    += e * h[dst,:]   (global_atomic_add_f32, no-return)
//   rowsum[src]   += e              (lane 0)
// h (25.6 MB) and out (25.6 MB) are L2-resident on MI455X (192 MB L2), so the
// 0.8 GB of gathers + 0.4 GB of atomics stay on-chip.
// ---------------------------------------------------------------------------
__global__ __launch_bounds__(256) void gat_edge(const float* __restrict__ H,
                                                const long long* __restrict__ edge,
                                                const float* __restrict__ avec,
                                                float* __restrict__ out,
                                                float* __restrict__ rowsum) {
    const int lane = threadIdx.x & 31;
    const int wave = threadIdx.x >> 5;
    const long long e = (long long)blockIdx.x * 8 + wave;
    if (e >= NEDGES) return;

    const int src = (int)edge[e];
    const int dst = (int)edge[(long long)NEDGES + e];

    const float4 hs = *(const float4*)(H + (size_t)src * DFEAT + lane * 4);
    const float4 hd = *(const float4*)(H + (size_t)dst * DFEAT + lane * 4);
    const float4 av = *(const float4*)(avec + lane * 4);

    float s0 = hs.x + hd.x; s0 = (s0 > 0.f) ? s0 : ALPHA * s0;
    float s1 = hs.y + hd.y; s1 = (s1 > 0.f) ? s1 : ALPHA * s1;
    float s2 = hs.z + hd.z; s2 = (s2 > 0.f) ? s2 : ALPHA * s2;
    float s3 = hs.w + hd.w; s3 = (s3 > 0.f) ? s3 : ALPHA * s3;

    float partial = s0 * av.x + s1 * av.y + s2 * av.z + s3 * av.w;
#pragma unroll
    for (int off = 16; off > 0; off >>= 1)
        partial += __shfl_xor(partial, off, 32);

    const float ee = __expf(partial);

    float* o = out + (size_t)src * DFEAT + lane * 4;
    atomicAdd(o + 0, ee * hd.x);
    atomicAdd(o + 1, ee * hd.y);
    atomicAdd(o + 2, ee * hd.z);
    atomicAdd(o + 3, ee * hd.w);
    if (lane == 0) atomicAdd(rowsum + src, ee);
}

// ---------------------------------------------------------------------------
// Kernel 3: h_prime = relu(h_prime / rowsum)
// ---------------------------------------------------------------------------
__global__ __launch_bounds__(256) void gat_finalize(float* __restrict__ out,
                                                    const float* __restrict__ rowsum) {
    const int i = blockIdx.x * blockDim.x + threadIdx.x;
    if (i >= NNODES * DFEAT) return;
    const float r = rowsum[i >> 7];   // i / DFEAT
    const float v = out[i] / r;
    out[i] = (v > 0.f) ? v : 0.f;
}

extern "C" void kernel_launch(void* const* d_in, const int* in_sizes, int n_in,
                              void* d_out, int out_size, void* d_ws, size_t ws_size,
                              hipStream_t stream) {
    const float*     X    = (const float*)d_in[0];       // [N, D]
    const long long* edge = (const long long*)d_in[1];   // [2, E] int64
    const float*     W    = (const float*)d_in[2];       // [D, D]
    const float*     avec = (const float*)d_in[3];       // [1, D]
    float*           out  = (float*)d_out;               // [N, D]

    float* H      = (float*)d_ws;                        // [N, D] projected feats
    float* rowsum = H + (size_t)NNODES * DFEAT;          // [N]

    // Zero accumulators (graph-capturable memset nodes).
    hipMemsetAsync(out,    0, (size_t)NNODES * DFEAT * sizeof(float), stream);
    hipMemsetAsync(rowsum, 0, (size_t)NNODES * sizeof(float), stream);

    // GEMM: 3125 row tiles, 8 waves (tiles) per 256-thread block.
    const int rowTiles  = NNODES / 16;                   // 3125
    const int gemmBlks  = (rowTiles + 7) / 8;            // 391
    gat_gemm_wmma<<<gemmBlks, 256, 0, stream>>>(X, W, H);

    // Edge phase: 1 wave per edge, 8 waves per block.
    const int edgeBlks = (NEDGES + 7) / 8;               // 100000
    gat_edge<<<edgeBlks, 256, 0, stream>>>(H, edge, avec, out, rowsum);

    // Finalize.
    const int finBlks = (NNODES * DFEAT + 255) / 256;    // 25000
    gat_finalize<<<finBlks, 256, 0, stream>>>(out, rowsum);
}